// VectorQuantizer_51556787421594
// MI455X (gfx1250) — compile-verified
//
#include <hip/hip_runtime.h>
#include <hip/hip_bf16.h>
#include <stddef.h>

// ---------------- problem constants ----------------
#define NROWS   16384      // flat rows (B*D*H*W / D)
#define DIM     256        // embedding_dim
#define KCB     8192       // num_embeddings
#define QELEMS  4194304    // NROWS*DIM
#define ENCELEMS 134217728 // NROWS*KCB
// d_out layout: [0, QELEMS) quantized_st | [QELEMS] vq_loss | [QELEMS+1 ...) encodings

typedef float v2f __attribute__((ext_vector_type(2)));
typedef float v8f __attribute__((ext_vector_type(8)));

// ---------------- ws layout (floats) ----------------
#define WS_ENORM   0
#define WS_EMBT    (8192)
#define WS_BESTV   (8192 + 2097152)
#define WS_BESTI   (WS_BESTV + 65536)
#define WS_IDXF    (WS_BESTI + 65536)
#define WS_PART    (WS_IDXF + 16384)

// ============ 1) zero encodings + loss slot ============
__global__ void vq_zero_kernel(float* __restrict__ out) {
    float4* p = (float4*)(out + QELEMS);          // 16B aligned (QELEMS % 4 == 0)
    const size_t n4 = (size_t)(ENCELEMS + 1) / 4; // 33554432 float4s
    const float4 z = make_float4(0.f, 0.f, 0.f, 0.f);
    size_t i = (size_t)blockIdx.x * blockDim.x + threadIdx.x;
    const size_t stride = (size_t)gridDim.x * blockDim.x;
    for (; i < n4; i += stride) p[i] = z;
    if (blockIdx.x == 0 && threadIdx.x == 0)
        out[(size_t)QELEMS + ENCELEMS] = 0.f;     // tail element
}

// ============ 2) ||e_k||^2 + transpose embedding ============
__global__ void vq_enorm_transpose_kernel(const float* __restrict__ emb,
                                          float* __restrict__ enorm,
                                          float* __restrict__ embT) {
    const int k = blockIdx.x * blockDim.x + threadIdx.x;   // 0..8191, coalesced reads
    float s = 0.f;
    float* dstT = embT + (size_t)k * DIM;
    for (int d = 0; d < DIM; ++d) {
        float v = emb[(size_t)d * KCB + k];
        s += v * v;
        dstT[d] = v;
    }
    enorm[k] = s;
}

// ============ 3) WMMA distance + fused argmin (double-buffered) ============
// grid = 512 blocks (128 row-blocks x 4 K-chunks), 256 threads (8 waves)
// wave owns 16 rows; WG shares B tiles (256x16) via ping-pong LDS.
__global__ __launch_bounds__(256, 1)
void vq_argmin_kernel(const float* __restrict__ inp,
                      const float* __restrict__ emb,
                      const float* __restrict__ enorm,
                      float* __restrict__ bestv,
                      int*   __restrict__ besti) {
    __shared__ float lds0[4096];   // 16 KB, pairs (B[2p][n], B[2p+1][n])
    __shared__ float lds1[4096];   // 16 KB, ping-pong buffer

    const int tid  = threadIdx.x;
    const int wave = tid >> 5;
    const int lane = tid & 31;
    const int ln   = lane & 15;    // N column within tile / M row selector
    const int hi   = lane >> 4;    // half-wave selector
    const int chunk  = blockIdx.x & 3;        // K-chunk of 128 n-tiles
    const int rowblk = blockIdx.x >> 2;       // 0..127
    const int base   = rowblk * 128 + wave * 16;

    // ---- A fragments (16x256) held in registers for the whole kernel ----
    // ISA 32-bit A 16x4 layout: lanes 0-15 -> M, VGPR0 = K=0|K=2, VGPR1 = K=1|K=3
    const int m = base + ln;
    const float* arow = inp + (size_t)m * DIM + 2 * hi;
    v2f a[64];
#pragma unroll
    for (int t = 0; t < 64; ++t) {
        a[t].x = arow[4 * t];
        a[t].y = arow[4 * t + 1];
    }

    float best[8];
    int   bidx[8];
#pragma unroll
    for (int v = 0; v < 8; ++v) { best[v] = 3.4e38f; bidx[v] = 0; }

    float4 vv[4];   // prefetch registers for the next B tile

    // fetch a B tile (256x16) into registers: 4 float4s / thread, coalesced
    auto fetch = [&](int nt) {
        const int n0 = (chunk * 128 + nt) * 16;
#pragma unroll
        for (int i = 0; i < 4; ++i) {
            const int lin = i * 256 + tid;           // float4 index 0..1023
            const int d   = lin >> 2;
            const int nq  = lin & 3;
            vv[i] = *(const float4*)(emb + (size_t)d * KCB + n0 + nq * 4);
        }
    };
    // deposit prefetched tile into LDS in WMMA-B pair layout
    auto stor = [&](float* buf) {
#pragma unroll
        for (int i = 0; i < 4; ++i) {
            const int lin = i * 256 + tid;
            const int d   = lin >> 2;
            const int nq  = lin & 3;
            const int pb  = ((d >> 1) * 16 + nq * 4) * 2 + (d & 1);
            buf[pb + 0] = vv[i].x;
            buf[pb + 2] = vv[i].y;
            buf[pb + 4] = vv[i].z;
            buf[pb + 6] = vv[i].w;
        }
    };
    // 64 WMMA k-steps on one tile, two independent accumulation chains
    auto compute = [&](const float* buf, int nt) {
        const int   col = (chunk * 128 + nt) * 16 + ln;
        const float en  = enorm[col];
        const float* bptr = buf + (hi * 16 + ln) * 2;
        v8f c0 = {}, c1 = {};
#pragma unroll
        for (int t = 0; t < 64; t += 2) {
            v2f b0 = *(const v2f*)(bptr + 64 * t);        // pair 2t+hi
            v2f b1 = *(const v2f*)(bptr + 64 * t + 64);
            c0 = __builtin_amdgcn_wmma_f32_16x16x4_f32(
                     false, a[t],     false, b0, (short)0, c0, false, false);
            c1 = __builtin_amdgcn_wmma_f32_16x16x4_f32(
                     false, a[t + 1], false, b1, (short)0, c1, false, false);
        }
        // running argmin of (||e||^2 - 2 x.e); ||x||^2 constant per row -> dropped
#pragma unroll
        for (int v = 0; v < 8; ++v) {
            const float sc = en - 2.0f * (c0[v] + c1[v]);
            if (sc < best[v]) { best[v] = sc; bidx[v] = col; }  // strict <: first wins
        }
    };

    // ---- software pipeline: prefetch tile nt+1 under compute of tile nt ----
    fetch(0);
    stor(lds0);
    __syncthreads();
    for (int j = 0; j < 64; ++j) {
        const int tA = 2 * j, tB = 2 * j + 1;
        fetch(tB);                 // loads in flight during compute below
        compute(lds0, tA);
        stor(lds1);
        __syncthreads();           // lds1 ready; everyone done with lds0
        if (tB < 127) fetch(tB + 1);
        compute(lds1, tB);
        if (tB < 127) stor(lds0);
        __syncthreads();           // lds0 ready; everyone done with lds1
    }

    // C/D layout: VGPR v, lanes 0-15 -> M=v, lanes 16-31 -> M=8+v; N = ln
#pragma unroll
    for (int v = 0; v < 8; ++v) {
        float val = best[v];
        int   id  = bidx[v];
        for (int off = 8; off >= 1; off >>= 1) {
            const float ov = __shfl_xor(val, off, 16);
            const int   oi = __shfl_xor(id,  off, 16);
            if (ov < val || (ov == val && oi < id)) { val = ov; id = oi; }
        }
        if (ln == 0) {
            const int row = base + v + 8 * hi;
            bestv[chunk * NROWS + row] = val;
            besti[chunk * NROWS + row] = id;
        }
    }
}

// ============ 4) merge 4 K-chunk partials ============
__global__ void vq_merge_kernel(const float* __restrict__ bestv,
                                const int*   __restrict__ besti,
                                int*         __restrict__ idxf) {
    const int mrow = blockIdx.x * blockDim.x + threadIdx.x;  // 0..16383
    float bv = bestv[mrow];
    int   bi = besti[mrow];
    for (int ch = 1; ch < 4; ++ch) {
        const float v = bestv[ch * NROWS + mrow];
        const int   i = besti[ch * NROWS + mrow];
        if (v < bv) { bv = v; bi = i; }   // strict <: earlier chunk (lower idx) wins ties
    }
    idxf[mrow] = bi;
}

// ============ 5) gather quantized + one-hot + SSE partials ============
// grid = 2048 blocks x 256 threads; one wave per row
__global__ __launch_bounds__(256)
void vq_gather_kernel(const float* __restrict__ inp,
                      const float* __restrict__ embT,
                      const int*   __restrict__ idxf,
                      float* __restrict__ out,        // d_out base
                      float* __restrict__ partials) {
    __shared__ float wsum[8];
    const int tid  = threadIdx.x;
    const int wave = tid >> 5;
    const int lane = tid & 31;
    const int mrow = blockIdx.x * 8 + wave;
    const int id   = idxf[mrow];

    const float* q = embT + (size_t)id * DIM;      // coalesced (transposed copy)
    const float* x = inp  + (size_t)mrow * DIM;
    float* outq    = out  + (size_t)mrow * DIM;

    float acc = 0.f;
#pragma unroll
    for (int j = 0; j < 8; ++j) {
        const int d = j * 32 + lane;
        const float qv = q[d];
        const float dv = qv - x[d];
        outq[d] = qv;                               // straight-through == quantized
        acc += dv * dv;
    }
    for (int off = 16; off >= 1; off >>= 1)
        acc += __shfl_xor(acc, off, 32);
    if (lane == 0) {
        wsum[wave] = acc;
        out[(size_t)QELEMS + 1 + (size_t)mrow * KCB + id] = 1.0f;  // one-hot
    }
    __syncthreads();
    if (tid == 0) {
        float s = 0.f;
        for (int w = 0; w < 8; ++w) s += wsum[w];   // fixed order -> deterministic
        partials[blockIdx.x] = s;
    }
}

// ============ 6) deterministic loss reduction ============
__global__ void vq_loss_kernel(const float* __restrict__ partials,
                               float* __restrict__ out) {
    __shared__ float red[256];
    const int tid = threadIdx.x;
    float s = 0.f;
    for (int i = tid; i < 2048; i += 256) s += partials[i];  // fixed order
    red[tid] = s;
    __syncthreads();
    for (int st = 128; st > 0; st >>= 1) {
        if (tid < st) red[tid] += red[tid + st];
        __syncthreads();
    }
    if (tid == 0) {
        // q_latent + 0.25*e_latent; numerically both equal mean((q-x)^2)
        out[QELEMS] = red[0] * 1.25f / (float)QELEMS;
    }
}

// ---------------- host launcher ----------------
extern "C" void kernel_launch(void* const* d_in, const int* in_sizes, int n_in,
                              void* d_out, int out_size, void* d_ws, size_t ws_size,
                              hipStream_t stream) {
    (void)in_sizes; (void)n_in; (void)out_size; (void)ws_size;
    const float* inp = (const float*)d_in[0];   // [16384, 256] flat
    const float* emb = (const float*)d_in[1];   // [256, 8192]
    float* out = (float*)d_out;

    float* ws       = (float*)d_ws;
    float* enorm    = ws + WS_ENORM;
    float* embT     = ws + WS_EMBT;
    float* bestv    = ws + WS_BESTV;
    int*   besti    = (int*)(ws + WS_BESTI);
    int*   idxf     = (int*)(ws + WS_IDXF);
    float* partials = ws + WS_PART;

    vq_zero_kernel<<<1024, 256, 0, stream>>>(out);
    vq_enorm_transpose_kernel<<<KCB / 256, 256, 0, stream>>>(emb, enorm, embT);
    vq_argmin_kernel<<<512, 256, 0, stream>>>(inp, emb, enorm, bestv, besti);
    vq_merge_kernel<<<NROWS / 256, 256, 0, stream>>>(bestv, besti, idxf);
    vq_gather_kernel<<<NROWS / 8, 256, 0, stream>>>(inp, embT, idxf, out, partials);
    vq_loss_kernel<<<1, 256, 0, stream>>>(partials, out);
}